// RegionProposalNetwork_59399397703950
// MI455X (gfx1250) — compile-verified
//
#include <hip/hip_runtime.h>
#include <hip/hip_bf16.h>

// ---------------- problem constants ----------------
#define N_IMG       8
#define A_CNT       960000
#define PRE_NMS_K   2000
#define CAP         2048          // padded sort capacity (power of two >= PRE_NMS_K)
#define POST_NMS_K  1000
#define NBINS       1024
#define IMG_W_F     1280.0f
#define IMG_H_F     800.0f
#define NMS_TH      0.7f
#define MIN_SZ      1e-3f
#define BBOX_CLIP   4.135166556742356f   // log(1000/16)

// ---------------- workspace layout (bytes) ----------------
// hist:   N_IMG * NBINS * 4              = 32768
// ctrs:   N_IMG * 4 u32 {t,quota,nsel,q} = 128
// keys:   N_IMG * CAP  * 8               = 131072
// boxes:  N_IMG * CAP  * 16              = 262144
#define OFF_HIST  ((size_t)0)
#define OFF_CTRS  ((size_t)32768)
#define OFF_KEYS  ((size_t)33024)
#define OFF_BOX   ((size_t)(33024 + 131072))

// ---------------- CDNA5 async global->LDS path ----------------
#ifndef __has_builtin
#define __has_builtin(x) 0
#endif

#if __has_builtin(__builtin_amdgcn_global_load_async_to_lds_b64) && \
    __has_builtin(__builtin_amdgcn_global_load_async_to_lds_b128) && \
    __has_builtin(__builtin_amdgcn_s_wait_asynccnt)
#define USE_ASYNC_LDS 1
#else
#define USE_ASYNC_LDS 0
#endif

#if USE_ASYNC_LDS
// Builtin signature (from clang diagnostic): pointee is an int vector.
typedef int v2i __attribute__((vector_size(2 * sizeof(int))));
typedef int v4i __attribute__((vector_size(4 * sizeof(int))));
typedef __attribute__((address_space(1))) v2i* g_v2i_p;
typedef __attribute__((address_space(3))) v2i* l_v2i_p;
typedef __attribute__((address_space(1))) v4i* g_v4i_p;
typedef __attribute__((address_space(3))) v4i* l_v4i_p;
#endif

__device__ __forceinline__ void ld_async_b64(const void* g, void* l) {
#if USE_ASYNC_LDS
  __builtin_amdgcn_global_load_async_to_lds_b64((g_v2i_p)g, (l_v2i_p)l, 0, 0);
#else
  *(unsigned long long*)l = *(const unsigned long long*)g;
#endif
}

__device__ __forceinline__ void ld_async_b128(const void* g, void* l) {
#if USE_ASYNC_LDS
  __builtin_amdgcn_global_load_async_to_lds_b128((g_v4i_p)g, (l_v4i_p)l, 0, 0);
#else
  *(float4*)l = *(const float4*)g;
#endif
}

__device__ __forceinline__ void wait_async_lds() {
#if USE_ASYNC_LDS
  __builtin_amdgcn_s_wait_asynccnt(0);
#endif
}

// order-preserving float -> uint key (larger float => larger key)
__device__ __forceinline__ unsigned int fkey(float f) {
  unsigned int u = __float_as_uint(f);
  return (u & 0x80000000u) ? ~u : (u | 0x80000000u);
}
__device__ __forceinline__ float fkey_inv(unsigned int k) {
  unsigned int u = (k & 0x80000000u) ? (k & 0x7FFFFFFFu) : ~k;
  return __uint_as_float(u);
}

// ---------------- kernel 1: zero workspace state ----------------
__global__ void rpn_init_kernel(unsigned int* hist, unsigned int* ctrs,
                                unsigned long long* keys) {
  int i = blockIdx.x * blockDim.x + threadIdx.x;
  if (i < N_IMG * NBINS) hist[i] = 0u;
  if (i < N_IMG * 4)     ctrs[i] = 0u;
  if (i < N_IMG * CAP)   keys[i] = 0ull;
}

// ---------------- kernel 2: per-image score-key histogram ----------------
__global__ void rpn_hist_kernel(const float* __restrict__ obj,
                                unsigned int* __restrict__ hist) {
  __shared__ unsigned int lh[NBINS];
  const int img = blockIdx.y;
  for (int i = threadIdx.x; i < NBINS; i += blockDim.x) lh[i] = 0u;
  __syncthreads();

  const float* p = obj + (size_t)img * A_CNT;
  const int stride = gridDim.x * blockDim.x;
  for (int i = blockIdx.x * blockDim.x + threadIdx.x; i < A_CNT; i += stride) {
    if (i + stride < A_CNT) __builtin_prefetch(&p[i + stride], 0, 3); // global_prefetch_b8
    atomicAdd(&lh[fkey(p[i]) >> 22], 1u);
  }
  __syncthreads();
  unsigned int* gh = hist + img * NBINS;
  for (int i = threadIdx.x; i < NBINS; i += blockDim.x)
    if (lh[i]) atomicAdd(&gh[i], lh[i]);
}

// ---------------- kernel 3: radix-select threshold bin ----------------
__global__ void rpn_thresh_kernel(const unsigned int* __restrict__ hist,
                                  unsigned int* __restrict__ ctrs) {
  const int img = blockIdx.x;
  if (threadIdx.x != 0) return;
  const unsigned int* h = hist + img * NBINS;
  unsigned int acc = 0, t = 0, quota = 0x7FFFFFFFu;
  for (int b = NBINS - 1; b >= 0; --b) {
    unsigned int c = h[b];
    if (acc + c >= PRE_NMS_K) { t = (unsigned int)b; quota = PRE_NMS_K - acc; break; }
    acc += c;
  }
  unsigned int* c4 = ctrs + img * 4;
  c4[0] = t; c4[1] = quota;  // c4[2]=nsel, c4[3]=in-bin counter already zero
}

// ---------------- kernel 4: select top-K, gather + decode boxes ----------------
__global__ void rpn_select_kernel(const float* __restrict__ obj,
                                  const float4* __restrict__ deltas,
                                  const float4* __restrict__ anchors,
                                  unsigned int* __restrict__ ctrs,
                                  unsigned long long* __restrict__ keys,
                                  float4* __restrict__ boxes) {
  const int img = blockIdx.y;
  unsigned int* c4 = ctrs + img * 4;
  const unsigned int t = c4[0], quota = c4[1];
  const float* p = obj + (size_t)img * A_CNT;
  const int stride = gridDim.x * blockDim.x;

  for (int i = blockIdx.x * blockDim.x + threadIdx.x; i < A_CNT; i += stride) {
    if (i + stride < A_CNT) __builtin_prefetch(&p[i + stride], 0, 3);
    const float logit = p[i];
    const unsigned int key = fkey(logit);
    const unsigned int bin = key >> 22;
    if (bin < t) continue;
    if (bin == t) {
      unsigned int q = atomicAdd(&c4[3], 1u);
      if (q >= quota) continue;
    }
    unsigned int slot = atomicAdd(&c4[2], 1u);
    if (slot >= CAP) continue;  // safety (cannot trigger: exactly PRE_NMS_K selected)

    // decode (torchvision BoxCoder, weights 1,1,1,1)
    const float4 an = anchors[i];
    const float4 dl = deltas[(size_t)img * A_CNT + i];
    const float w  = an.z - an.x, h = an.w - an.y;
    const float cx = an.x + 0.5f * w, cy = an.y + 0.5f * h;
    const float dw = fminf(dl.z, BBOX_CLIP), dh = fminf(dl.w, BBOX_CLIP);
    const float pcx = dl.x * w + cx, pcy = dl.y * h + cy;
    const float pw = expf(dw) * w, ph = expf(dh) * h;
    float x1 = pcx - 0.5f * pw, y1 = pcy - 0.5f * ph;
    float x2 = pcx + 0.5f * pw, y2 = pcy + 0.5f * ph;
    x1 = fminf(fmaxf(x1, 0.0f), IMG_W_F); x2 = fminf(fmaxf(x2, 0.0f), IMG_W_F);
    y1 = fminf(fmaxf(y1, 0.0f), IMG_H_F); y2 = fminf(fmaxf(y2, 0.0f), IMG_H_F);
    const bool valid = ((x2 - x1) >= MIN_SZ) && ((y2 - y1) >= MIN_SZ);

    const float score = 1.0f / (1.0f + expf(-logit));          // sigmoid
    const float skey  = valid ? score : -__builtin_huge_valf();
    // descending u64 sort key; ties resolved toward lower slot (stable argsort)
    keys[img * CAP + slot] =
        ((unsigned long long)fkey(skey) << 32) | (unsigned long long)(0xFFFFFFFFu - slot);
    boxes[img * CAP + slot] = make_float4(x1, y1, x2, y2);
  }
}

// ---------------- kernel 5: per-image sort + NMS + compaction ----------------
__global__ __launch_bounds__(1024) void rpn_sort_nms_kernel(
    const unsigned long long* __restrict__ keys,
    const float4* __restrict__ boxes,
    float* __restrict__ out) {
  __shared__ unsigned long long sKeys[CAP];   // 16 KB (aliased as scan space later)
  __shared__ float4 sBox[CAP];                // 32 KB
  __shared__ float sArea[CAP];                // 8 KB (reused for scores after NMS)
  __shared__ unsigned char sKeep[CAP];        // 2 KB

  const int img = blockIdx.x;
  const int tid = threadIdx.x;
  const int nt  = blockDim.x;
  const unsigned long long* gk = keys + img * CAP;

  // ---- phase A: async-stage sort keys into LDS ----
  for (int i = tid; i < CAP; i += nt) ld_async_b64(&gk[i], &sKeys[i]);
  wait_async_lds();
  __syncthreads();

  // ---- phase B: bitonic sort, descending on u64 ----
  for (int k = 2; k <= CAP; k <<= 1) {
    for (int j = k >> 1; j > 0; j >>= 1) {
      __syncthreads();
      for (int i = tid; i < CAP; i += nt) {
        const int ixj = i ^ j;
        if (ixj > i) {
          const unsigned long long a = sKeys[i], b = sKeys[ixj];
          const bool up = ((i & k) == 0);
          if (up ? (a < b) : (a > b)) { sKeys[i] = b; sKeys[ixj] = a; }
        }
      }
    }
  }
  __syncthreads();

  // ---- phase C: async-gather boxes in sorted order; init keep flags ----
  for (int i = tid; i < CAP; i += nt) {
    const unsigned long long kv = sKeys[i];
    const unsigned int sk   = (unsigned int)(kv >> 32);
    const unsigned int slot = (kv == 0ull) ? 0u : (0xFFFFFFFFu - (unsigned int)kv);
    sKeep[i] = (kv != 0ull && sk > 0x007FFFFFu) ? 1 : 0;  // 0x007FFFFF == key(-inf)
    ld_async_b128(&boxes[(size_t)img * CAP + slot], &sBox[i]);
  }
  wait_async_lds();
  __syncthreads();

  // ---- phase D: areas ----
  for (int i = tid; i < CAP; i += nt) {
    const float4 b = sBox[i];
    sArea[i] = (b.z - b.x) * (b.w - b.y);
  }
  __syncthreads();

  // ---- phase E: greedy sequential NMS (barrier only on kept rows) ----
  for (int i = 0; i < CAP; ++i) {
    if (!sKeep[i]) continue;        // uniform read; no writes since last barrier
    const float4 bi = sBox[i];
    const float ai  = sArea[i];
    for (int j = i + 1 + tid; j < CAP; j += nt) {
      if (!sKeep[j]) continue;
      const float4 bj = sBox[j];
      const float xx1 = fmaxf(bi.x, bj.x), yy1 = fmaxf(bi.y, bj.y);
      const float xx2 = fminf(bi.z, bj.z), yy2 = fminf(bi.w, bj.w);
      const float iw = fmaxf(xx2 - xx1, 0.0f), ih = fmaxf(yy2 - yy1, 0.0f);
      const float inter = iw * ih;
      const float iou = inter / (ai + sArea[j] - inter);
      if (iou > NMS_TH) sKeep[j] = 0;
    }
    __syncthreads();
  }

  // ---- phase F: scores (reuse sArea), chunked prefix-sum, output ----
  for (int i = tid; i < CAP; i += nt) {
    const unsigned int sk = (unsigned int)(sKeys[i] >> 32);
    sArea[i] = fkey_inv(sk);        // kept rows always have a finite sigmoid score
  }
  __syncthreads();

  unsigned int* sScan  = (unsigned int*)sKeys;         // alias: 2048 u32 (first 8 KB)
  unsigned int* sChunk = ((unsigned int*)sKeys) + CAP; // alias: 256 u32 (next 1 KB)
  if (tid < 256) {
    unsigned int s = 0;
    for (int k = 0; k < 8; ++k) {
      const int idx = tid * 8 + k;
      sScan[idx] = s;
      s += sKeep[idx];
    }
    sChunk[tid] = s;
  }
  __syncthreads();
  if (tid == 0) {
    unsigned int s = 0;
    for (int c = 0; c < 256; ++c) { const unsigned int v = sChunk[c]; sChunk[c] = s; s += v; }
  }
  __syncthreads();

  float* o = out + (size_t)img * POST_NMS_K * 5;
  for (int i = tid; i < POST_NMS_K * 5; i += nt) o[i] = 0.0f;
  __syncthreads();
  for (int i = tid; i < CAP; i += nt) {
    if (sKeep[i]) {
      const unsigned int r = sChunk[i >> 3] + sScan[i];
      if (r < POST_NMS_K) {
        const float4 b = sBox[i];
        o[r * 5 + 0] = b.x; o[r * 5 + 1] = b.y;
        o[r * 5 + 2] = b.z; o[r * 5 + 3] = b.w;
        o[r * 5 + 4] = sArea[i];
      }
    }
  }
}

// ---------------- host entry ----------------
extern "C" void kernel_launch(void* const* d_in, const int* in_sizes, int n_in,
                              void* d_out, int out_size, void* d_ws, size_t ws_size,
                              hipStream_t stream) {
  const float*  obj     = (const float*)d_in[0];   // [8, 960000]
  const float4* deltas  = (const float4*)d_in[1];  // [8, 960000, 4]
  const float4* anchors = (const float4*)d_in[2];  // [960000, 4]
  float* out = (float*)d_out;                      // [8, 1000, 5]

  char* ws = (char*)d_ws;
  unsigned int*       hist  = (unsigned int*)(ws + OFF_HIST);
  unsigned int*       ctrs  = (unsigned int*)(ws + OFF_CTRS);
  unsigned long long* keys  = (unsigned long long*)(ws + OFF_KEYS);
  float4*             boxes = (float4*)(ws + OFF_BOX);

  rpn_init_kernel<<<64, 256, 0, stream>>>(hist, ctrs, keys);
  rpn_hist_kernel<<<dim3(128, N_IMG), 256, 0, stream>>>(obj, hist);
  rpn_thresh_kernel<<<N_IMG, 64, 0, stream>>>(hist, ctrs);
  rpn_select_kernel<<<dim3(128, N_IMG), 256, 0, stream>>>(obj, deltas, anchors,
                                                          ctrs, keys, boxes);
  rpn_sort_nms_kernel<<<N_IMG, 1024, 0, stream>>>(keys, boxes, out);
}